// PositionalEncodeing_7919919693945
// MI455X (gfx1250) — compile-verified
//
#include <hip/hip_runtime.h>
#include <math.h>

#define PE_B 4
#define PE_S 8192
#define PE_D 1024
#define PE_HALF (PE_D / 2)   // 512 angle pairs per row
#define PE_R 8               // rows (sequence positions) per block
#define THREADS 256          // 8 wave32 waves; each thread -> 2 column pairs

// ---------------------------------------------------------------------------
// Kernel 1: inv_freq[j] = base^(j/512), j in [0,512). Only 512 distinct powf
// values exist for the whole problem; hoisting them removes powf entirely
// from the hot kernel.
// ---------------------------------------------------------------------------
__global__ __launch_bounds__(PE_HALF)
void pe_invfreq_kernel(const int* __restrict__ basep, float* __restrict__ inv) {
    const int j = threadIdx.x;
    const float b = (float)basep[0];
    inv[j] = powf(b, (float)j * (1.0f / (float)PE_HALF));
}

// ---------------------------------------------------------------------------
// Kernel 2: one block per 8 consecutive sequence positions. Per thread:
//   - 2 columns (pairs j = 2t, 2t+1)
//   - precise sincosf for the base row angle and for the per-column step
//     angle f = 1/inv_freq, then 7 rotation steps (4 FMA-class ops/col/row)
//     instead of 7 more argument-reduced sincosf pairs  (~4x less VALU).
// Rows are staged in a 32 KiB LDS tile, then replicated to all 4 batch
// copies with async LDS->global B128 stores (ASYNCcnt path): the 4x batch
// replication never occupies VGPRs and drains behind other waves' VALU work.
// ---------------------------------------------------------------------------
__global__ __launch_bounds__(THREADS)
void pe_row_kernel(const int* __restrict__ basep,
                   const float* __restrict__ inv_tbl,   // nullptr -> inline powf
                   float* __restrict__ out) {
    __shared__ float rows[PE_R * PE_D];   // 32 KiB: 8 rows x 4 KiB

    const int s0 = blockIdx.x * PE_R;     // first sequence position of tile
    const int t  = threadIdx.x;
    const float pos = (float)s0;

    float inv0, inv1;
    if (inv_tbl != nullptr) {
        const float2 iv = ((const float2*)inv_tbl)[t];  // inv[2t], inv[2t+1]
        inv0 = iv.x;
        inv1 = iv.y;
    } else {
        const float b = (float)basep[0];
        inv0 = powf(b, (float)(2 * t)     * (1.0f / (float)PE_HALF));
        inv1 = powf(b, (float)(2 * t + 1) * (1.0f / (float)PE_HALF));
    }

    // Base-row angles (precise: |ang| reaches ~8184 rad, needs full
    // argument reduction to track jnp.sin/jnp.cos) and per-row step angles.
    const float f0 = 1.0f / inv0;         // angle increment per position
    const float f1 = 1.0f / inv1;
    const float a0 = pos / inv0;
    const float a1 = pos / inv1;

    float s0v, c0v, s1v, c1v, sf0, cf0, sf1, cf1;
    sincosf(a0, &s0v, &c0v);
    sincosf(a1, &s1v, &c1v);
    sincosf(f0, &sf0, &cf0);
    sincosf(f1, &sf1, &cf1);

#pragma unroll
    for (int k = 0; k < PE_R; ++k) {
        ((float4*)&rows[k * PE_D])[t] = make_float4(s0v, c0v, s1v, c1v);
        // rotate: (sin,cos)(a) -> (sin,cos)(a + f)
        const float ns0 = fmaf(s0v, cf0,  c0v * sf0);
        const float nc0 = fmaf(c0v, cf0, -s0v * sf0);
        const float ns1 = fmaf(s1v, cf1,  c1v * sf1);
        const float nc1 = fmaf(c1v, cf1, -s1v * sf1);
        s0v = ns0; c0v = nc0; s1v = ns1; c1v = nc1;
    }

    // s_wait_dscnt 0 + barrier: DS writes must be visible before the async
    // engine reads the LDS tile.
    __syncthreads();

    // LDS byte offset = low 32 bits of the generic (flat) pointer: the
    // AS3->flat cast is {aperture_hi32, lds_offset}.
    const unsigned lds_base = (unsigned)(size_t)&rows[0] + (unsigned)(16 * t);
    const unsigned col_byte = (unsigned)(16 * t);          // 4t floats

#pragma unroll
    for (int k = 0; k < PE_R; ++k) {
        const unsigned lds_off  = lds_base + (unsigned)(k * PE_D * 4);
        const unsigned row_byte = (unsigned)(s0 + k) * (PE_D * 4) + col_byte;
#pragma unroll
        for (int b = 0; b < PE_B; ++b) {
            const unsigned byte_off = row_byte + (unsigned)b * (PE_S * PE_D * 4);
            // GVS form: vaddr = u32 byte offset, vsrc = LDS byte address,
            // saddr = 64-bit base. Tracked by ASYNCcnt (<= 32 outstanding,
            // under the 63-entry counter limit).
            asm volatile("global_store_async_from_lds_b128 %0, %1, %2"
                         :
                         : "v"(byte_off), "v"(lds_off), "s"(out)
                         : "memory");
        }
    }

    // Drain our async stores before the block (and its LDS) retires.
    asm volatile("s_wait_asynccnt 0" ::: "memory");
}

// ---------------------------------------------------------------------------
// d_in[0] = x (4,8192,1024) f32 — unused (shape only)
// d_in[1] = base (int scalar)
// d_out   = (4,8192,1024) f32
// ---------------------------------------------------------------------------
extern "C" void kernel_launch(void* const* d_in, const int* in_sizes, int n_in,
                              void* d_out, int out_size, void* d_ws, size_t ws_size,
                              hipStream_t stream) {
    (void)in_sizes; (void)n_in; (void)out_size;

    const int* basep = (const int*)d_in[1];
    float* out = (float*)d_out;

    float* inv = (ws_size >= (size_t)PE_HALF * sizeof(float)) ? (float*)d_ws
                                                              : nullptr;
    if (inv != nullptr) {
        pe_invfreq_kernel<<<1, PE_HALF, 0, stream>>>(basep, inv);
    }
    pe_row_kernel<<<PE_S / PE_R, THREADS, 0, stream>>>(basep, inv, out);
}